// Network_60498909332067
// MI455X (gfx1250) — compile-verified
//
#include <hip/hip_runtime.h>

typedef float v2f __attribute__((ext_vector_type(2)));
typedef float v4f __attribute__((ext_vector_type(4)));
typedef float v8f __attribute__((ext_vector_type(8)));

#define N_PTS   131072
#define JNT     24
#define RCOMP   32
#define CDIM    32
#define G0      512
#define G1      512
#define G2      128

// workspace layout (float offsets)
#define WS_LT0   0
#define WS_LT1   (G0*128)              // 65536
#define WS_LT2   (WS_LT1 + G1*128)     // 131072
#define WS_BSTR  (WS_LT2 + G2*128)     // 147456
#define WS_TOTAL (WS_BSTR + 3*32*128)  // 159744 floats = 639 KB

// ---------------------------------------------------------------------------
// Prep kernel:
//  (a) lineT[a][g][k*32+r] = coord_line_a[topk_id[k]][r][g]   (contiguous
//      512B per (g) sample in the main kernel's gather)
//  (b) B-fragment stream for V_WMMA_F32_16X16X4_F32, topk_weight folded in.
//      KR permutation: K-local {0,1,2,3} of step s <-> kr = h*64 + 2s + j
//      (j = B VGPR index, h = lane half). Per (plane,step) each lane stores a
//      16B quad {nt0.v0, nt0.v1, nt1.v0, nt1.v1} at [..+4*lane], so the main
//      kernel fetches BOTH N-tiles' B fragments with a single ds_load_b128.
// ---------------------------------------------------------------------------
__global__ __launch_bounds__(256)
void cdna5_prep_kernel(const float* __restrict__ cl0,
                       const float* __restrict__ cl1,
                       const float* __restrict__ cl2,
                       const float* __restrict__ fl0,
                       const float* __restrict__ fl1,
                       const float* __restrict__ fl2,
                       const float* __restrict__ tw,
                       const int*   __restrict__ tid,
                       float* __restrict__ ws)
{
    int t = blockIdx.x * 256 + threadIdx.x;
    if (t >= WS_TOTAL) return;
    if (t < WS_BSTR) {
        const float* cl; int G; int local = t;
        if (local < WS_LT1)      {                  cl = cl0; G = G0; }
        else if (local < WS_LT2) { local -= WS_LT1; cl = cl1; G = G1; }
        else                     { local -= WS_LT2; cl = cl2; G = G2; }
        int g  = local >> 7;
        int kr = local & 127;
        int k = kr >> 5, r = kr & 31;
        int pid = tid[k];
        ws[t] = cl[(pid * RCOMP + r) * G + g];
    } else {
        int u    = t - WS_BSTR;        // (p*32+s)*128 + j
        int j    = u & 127;
        int ps   = u >> 7;             // p*32 + s
        int s    = ps & 31;
        int p    = ps >> 5;
        int lane = j >> 2;
        int quad = j & 3;              // {nt0.v0, nt0.v1, nt1.v0, nt1.v1}
        int nt2  = quad >> 1;
        int pair = quad & 1;
        int hh   = lane >> 4;
        int l    = lane & 15;
        int kr   = hh * 64 + 2 * s + pair;
        int k = kr >> 5, r = kr & 31;
        int c = nt2 * 16 + l;
        const float* fl = (p == 0) ? fl0 : ((p == 1) ? fl1 : fl2);
        int pid = tid[k];
        ws[t] = tw[k] * fl[(pid * RCOMP + r) * CDIM + c];
    }
}

// ---------------------------------------------------------------------------
// Fused kernel: LBS (pointwise) + tri-plane sampling + f32 WMMA feature GEMM.
// 128 threads = 4 waves; each wave owns two 16-point tiles (its own points).
// waves_per_eu(4): 256-VGPR budget, 4 waves/SIMD to hide L2 gather latency.
// tile and chunk loops kept rolled so live state (~225 VGPRs) fits the budget
// without spilling B fragments to scratch.
// ---------------------------------------------------------------------------
__global__ __launch_bounds__(128, 4)
__attribute__((amdgpu_waves_per_eu(4)))
void cdna5_fused_kernel(const float* __restrict__ pose_pts,
                        const float* __restrict__ pose_dirs,
                        const float* __restrict__ skin_w,
                        const float* __restrict__ A,
                        const float* __restrict__ bigA,
                        const float* __restrict__ tb,
                        const float* __restrict__ ws,
                        float* __restrict__ out)
{
    __shared__ float s_b[3 * 32 * 128];      // 48 KB B-fragment stream
    __shared__ float s_npts[128][3];

    const int t = threadIdx.x;
    const int n = blockIdx.x * 128 + t;

    // stage B stream into LDS (96 floats per thread, b128 copies)
    {
        const float* bsrc = ws + WS_BSTR;
        #pragma unroll
        for (int i = 0; i < 24; ++i) {
            int idx = (i * 128 + t) * 4;
            *(v4f*)&s_b[idx] = *(const v4f*)&bsrc[idx];
        }
    }

    // -------- phase 1: blend transforms, invert, LBS --------
    float TA[12], TB[12];
    #pragma unroll
    for (int i = 0; i < 12; ++i) { TA[i] = 0.f; TB[i] = 0.f; }
    for (int j = 0; j < JNT; ++j) {
        float wj = skin_w[n * JNT + j];
        const float* Aj = A + j * 16;     // rows 0..2, cols 0..3 = first 12
        const float* Bj = bigA + j * 16;
        #pragma unroll
        for (int i = 0; i < 12; ++i) {
            TA[i] = fmaf(wj, Aj[i], TA[i]);
            TB[i] = fmaf(wj, Bj[i], TB[i]);
        }
    }
    float a00=TA[0],a01=TA[1],a02=TA[2];
    float a10=TA[4],a11=TA[5],a12=TA[6];
    float a20=TA[8],a21=TA[9],a22=TA[10];
    float i00 = a11*a22 - a12*a21, i01 = a02*a21 - a01*a22, i02 = a01*a12 - a02*a11;
    float i10 = a12*a20 - a10*a22, i11 = a00*a22 - a02*a20, i12 = a02*a10 - a00*a12;
    float i20 = a10*a21 - a11*a20, i21 = a01*a20 - a00*a21, i22 = a00*a11 - a01*a10;
    float det  = a00*i00 + a01*i10 + a02*i20;
    float rdet = 1.0f / det;

    float px = pose_pts[n*3+0], py = pose_pts[n*3+1], pz = pose_pts[n*3+2];
    float dx = pose_dirs[n*3+0], dy = pose_dirs[n*3+1], dz = pose_dirs[n*3+2];
    float qx = px - TA[3], qy = py - TA[7], qz = pz - TA[11];
    float tx = (i00*qx + i01*qy + i02*qz) * rdet;
    float ty = (i10*qx + i11*qy + i12*qz) * rdet;
    float tz = (i20*qx + i21*qy + i22*qz) * rdet;
    float cx = TB[0]*tx + TB[1]*ty + TB[2]*tz + TB[3];
    float cy = TB[4]*tx + TB[5]*ty + TB[6]*tz + TB[7];
    float cz = TB[8]*tx + TB[9]*ty + TB[10]*tz + TB[11];
    float ux = (i00*dx + i01*dy + i02*dz) * rdet;
    float uy = (i10*dx + i11*dy + i12*dz) * rdet;
    float uz = (i20*dx + i21*dy + i22*dz) * rdet;
    float vx = TB[0]*ux + TB[1]*uy + TB[2]*uz;
    float vy = TB[4]*ux + TB[5]*uy + TB[6]*uz;
    float vz = TB[8]*ux + TB[9]*uy + TB[10]*uz;

    size_t ob = (size_t)n * 102;
    out[ob+0] = cx; out[ob+1] = cy; out[ob+2] = cz;
    out[ob+3] = vx; out[ob+4] = vy; out[ob+5] = vz;

    float lo0 = tb[0], lo1 = tb[1], lo2 = tb[2];
    float hi0 = tb[3], hi1 = tb[4], hi2 = tb[5];
    s_npts[t][0] = (cx - lo0) / (hi0 - lo0) * 2.f - 1.f;
    s_npts[t][1] = (cy - lo1) / (hi1 - lo1) * 2.f - 1.f;
    s_npts[t][2] = (cz - lo2) / (hi2 - lo2) * 2.f - 1.f;

    __syncthreads();

    // -------- phase 2+3: sample lines + WMMA GEMM --------
    const int lane = t & 31;
    const int wv   = t >> 5;
    const int pt   = lane & 15;   // point-in-tile (A rows M, and D column N)
    const int h    = lane >> 4;   // lane half -> which 64 of the 128 KR values

    const int Gs[3] = {G0, G1, G2};
    const float* lt[3] = { ws + WS_LT0, ws + WS_LT1, ws + WS_LT2 };

    #pragma unroll 1
    for (int tile = 2*wv; tile <= 2*wv + 1; ++tile) {
        const int ptLocal = tile * 16 + pt;
        const int nbase   = blockIdx.x * 128 + tile * 16;

        int   off0[3], off1[3];
        float wA[3], wB[3];
        #pragma unroll
        for (int a = 0; a < 3; ++a) {
            float x   = s_npts[ptLocal][a];
            float pos = (x + 1.0f) * 0.5f * (float)(Gs[a] - 1);
            float f   = floorf(pos);
            f = fminf(fmaxf(f, 0.0f), (float)(Gs[a] - 1));
            int ii0 = (int)f;
            int ii1 = (ii0 + 1 < Gs[a]) ? ii0 + 1 : Gs[a] - 1;
            float wf = pos - f;               // w vs clipped i0 (matches ref)
            off0[a] = ii0 * 128 + h * 64;
            off1[a] = ii1 * 128 + h * 64;
            wA[a] = 1.0f - wf;
            wB[a] = wf;
        }

        v8f acc[6];
        #pragma unroll
        for (int i = 0; i < 6; ++i) { v8f z = {0,0,0,0,0,0,0,0}; acc[i] = z; }

        #pragma unroll 1
        for (int chunk = 0; chunk < 4; ++chunk) {
            float ca[3][16];                  // this lane's 16 KR coords/axis
            #pragma unroll
            for (int a = 0; a < 3; ++a) {
                const float* b0p = lt[a] + off0[a] + chunk * 16;
                const float* b1p = lt[a] + off1[a] + chunk * 16;
                #pragma unroll
                for (int q = 0; q < 4; ++q) {
                    v4f v0 = *(const v4f*)(b0p + q * 4);
                    v4f v1 = *(const v4f*)(b1p + q * 4);
                    v4f vv = v0 * wA[a] + v1 * wB[a];
                    ca[a][q*4+0] = vv.x; ca[a][q*4+1] = vv.y;
                    ca[a][q*4+2] = vv.z; ca[a][q*4+3] = vv.w;
                }
            }
            #pragma unroll
            for (int p = 0; p < 3; ++p) {     // MAT_MODE (0,1),(0,2),(1,2)
                const int ax0 = (p == 2) ? 1 : 0;
                const int ax1 = (p == 0) ? 1 : 2;
                float prod[16];
                #pragma unroll
                for (int q = 0; q < 16; ++q) prod[q] = ca[ax0][q] * ca[ax1][q];
                #pragma unroll
                for (int s8 = 0; s8 < 8; ++s8) {
                    const int s = chunk * 8 + s8;
                    v2f av; av.x = prod[2*s8]; av.y = prod[2*s8+1];
                    const int boff = (p * 32 + s) * 128 + lane * 4;
                    v4f bq = *(const v4f*)&s_b[boff];
                    v2f b0; b0.x = bq.x; b0.y = bq.y;
                    v2f b1; b1.x = bq.z; b1.y = bq.w;
                    acc[p*2+0] = __builtin_amdgcn_wmma_f32_16x16x4_f32(
                        false, av, false, b0, (short)0, acc[p*2+0], false, false);
                    acc[p*2+1] = __builtin_amdgcn_wmma_f32_16x16x4_f32(
                        false, av, false, b1, (short)0, acc[p*2+1], false, false);
                }
            }
        }

        // D layout: VGPR v -> M = v + 8*h (point row), lane&15 -> N (feat col)
        #pragma unroll
        for (int p = 0; p < 3; ++p) {
            #pragma unroll
            for (int nt2 = 0; nt2 < 2; ++nt2) {
                v8f d = acc[p*2+nt2];
                const int col = 6 + p * 32 + nt2 * 16 + pt;
                #pragma unroll
                for (int v = 0; v < 8; ++v) {
                    out[(size_t)(nbase + v + 8*h) * 102 + col] = d[v];
                }
            }
        }
    }
}

extern "C" void kernel_launch(void* const* d_in, const int* in_sizes, int n_in,
                              void* d_out, int out_size, void* d_ws, size_t ws_size,
                              hipStream_t stream)
{
    const float* pose_pts  = (const float*)d_in[0];
    const float* pose_dirs = (const float*)d_in[1];
    const float* skin_w    = (const float*)d_in[2];
    const float* A         = (const float*)d_in[3];
    const float* bigA      = (const float*)d_in[4];
    const float* tb        = (const float*)d_in[5];
    const float* cl0       = (const float*)d_in[6];
    const float* cl1       = (const float*)d_in[7];
    const float* cl2       = (const float*)d_in[8];
    const float* fl0       = (const float*)d_in[9];
    const float* fl1       = (const float*)d_in[10];
    const float* fl2       = (const float*)d_in[11];
    const float* tw        = (const float*)d_in[12];
    const int*   tid       = (const int*)d_in[13];
    float* out = (float*)d_out;
    float* ws  = (float*)d_ws;   // needs 159744 floats (639 KB)

    (void)in_sizes; (void)n_in; (void)out_size; (void)ws_size;

    const int prepBlocks = (WS_TOTAL + 255) / 256;
    cdna5_prep_kernel<<<prepBlocks, 256, 0, stream>>>(cl0, cl1, cl2,
                                                      fl0, fl1, fl2,
                                                      tw, tid, ws);
    cdna5_fused_kernel<<<N_PTS / 128, 128, 0, stream>>>(pose_pts, pose_dirs,
                                                        skin_w, A, bigA, tb,
                                                        ws, out);
}